// VllmMixtureOfExpertsOp_71141838291315
// MI455X (gfx1250) — compile-verified
//
#include <hip/hip_runtime.h>
#include <hip/hip_bf16.h>
#include <stddef.h>

// Problem constants (match reference)
#define T_TOK  16384
#define H_DIM  2048
#define I_DIM  1024
#define E_EXP  8
#define K_TOP  2

#define MROWS  128                              // tokens per x-tile (register-blocked M)
#define MT     (MROWS / 16)                     // 8 m-subtiles per wave
#define NTMAX  (T_TOK * K_TOP + MROWS * E_EXP)  // padded entry capacity
#define BK     32                               // K-chunk per stage
#define LSTR   36                               // padded LDS row stride (floats): conflict-free, 16B-aligned rows

typedef __attribute__((ext_vector_type(2))) float v2f;
typedef __attribute__((ext_vector_type(4))) float v4f;
typedef __attribute__((ext_vector_type(8))) float v8f;
typedef __attribute__((ext_vector_type(4))) int   v4i;

typedef __attribute__((address_space(1))) v4i v4i_g;   // global int4
typedef __attribute__((address_space(3))) v4i v4i_l;   // LDS int4

// ---------------- gfx1250 async global->LDS staging ----------------
#if defined(__has_builtin)
#  if __has_builtin(__builtin_amdgcn_global_load_async_to_lds_b128) && \
      __has_builtin(__builtin_amdgcn_s_wait_asynccnt)
#    define USE_ASYNC_LDS 1
#  endif
#endif

__device__ __forceinline__ void cp16_to_lds(float* ldst, const float* gsrc) {
#ifdef USE_ASYNC_LDS
  __builtin_amdgcn_global_load_async_to_lds_b128(
      (v4i_g*)(void*)gsrc,
      (v4i_l*)(void*)ldst, 0, 0);
#else
  *(v4f*)ldst = *(const v4f*)gsrc;
#endif
}

__device__ __forceinline__ void wait_async_lds() {
#ifdef USE_ASYNC_LDS
  __builtin_amdgcn_s_wait_asynccnt(0);
#endif
}

// ---------------- workspace layout (bytes) ----------------
#define HDR_OFF   ((size_t)0)                                  // int offsets[0..8]
#define POS_OFF   ((size_t)256)                                // int pos[T*K]
#define ETOK_OFF  (POS_OFF + (size_t)T_TOK * K_TOP * 4)        // int etok[NTMAX]
#define ECOEF_OFF (ETOK_OFF + (size_t)NTMAX * 4)               // float ecoef[NTMAX]
#define ACT_OFF   (((ECOEF_OFF + (size_t)NTMAX * 4) + 1023) & ~(size_t)1023)
#define DOWN_OFF  (ACT_OFF + (size_t)NTMAX * I_DIM * 4)

// ---------------- router: build per-expert gathered token lists ----------------
__global__ __launch_bounds__(256) void moe_route(const int* __restrict__ rt,
                                                 const float* __restrict__ rw,
                                                 int* __restrict__ hdr,
                                                 int* __restrict__ pos,
                                                 int* __restrict__ etok,
                                                 float* __restrict__ ecoef) {
  __shared__ int cnt[E_EXP];
  __shared__ int base[E_EXP + 1];
  __shared__ int cur[E_EXP];
  const int tid = threadIdx.x;
  if (tid < E_EXP) { cnt[tid] = 0; cur[tid] = 0; }
  __syncthreads();
  for (int i = tid; i < T_TOK * K_TOP; i += 256) atomicAdd(&cnt[rt[i]], 1);
  __syncthreads();
  if (tid == 0) {
    int b = 0;
    for (int e = 0; e < E_EXP; ++e) { base[e] = b; b += (cnt[e] + (MROWS - 1)) & ~(MROWS - 1); }
    base[E_EXP] = b;
    for (int e = 0; e <= E_EXP; ++e) hdr[e] = base[e];
  }
  __syncthreads();
  const int total = base[E_EXP];
  for (int j = tid; j < total; j += 256) { etok[j] = 0; ecoef[j] = 0.0f; }
  __syncthreads();
  for (int i = tid; i < T_TOK * K_TOP; i += 256) {
    const int e = rt[i];
    const int p = base[e] + atomicAdd(&cur[e], 1);
    etok[p]  = i >> 1;            // K_TOP == 2
    ecoef[p] = rw[i];
    pos[i]   = p;
  }
}

// ---------------- kernel 2: gu = X_tile @ w13[e]^T ; act = silu(g)*u ----------------
// grid.x = entry tiles (128 tokens), grid.y = 8 (block covers 8 column-pairs, one per wave)
__global__ __launch_bounds__(256) void moe_gateup(const float* __restrict__ X,
                                                  const float* __restrict__ w13,
                                                  const int* __restrict__ hdr,
                                                  const int* __restrict__ etok,
                                                  float* __restrict__ act) {
  __shared__ float xs[MROWS * LSTR];            // 18432 B
  __shared__ float wsg[8][16 * LSTR];           // 18432 B
  __shared__ float wsu[8][16 * LSTR];           // 18432 B
  __shared__ int stok[MROWS];

  const int tile = blockIdx.x;
  const int tb = tile * MROWS;
  if (tb >= hdr[E_EXP]) return;                 // uniform early-exit (before barriers)
  int e = 0;
  #pragma unroll
  for (int i = 1; i < E_EXP; ++i) if (tb >= hdr[i]) e = i;

  const int tid = threadIdx.x;
  const int lane = tid & 31;
  const int wave = tid >> 5;
  if (tid < MROWS) stok[tid] = etok[tb + tid];
  __syncthreads();

  const int pair = blockIdx.y * 8 + wave;       // 0..63 : (g col-tile, u col-tile)
  const float* wg_base = w13 + ((size_t)e * (2 * I_DIM) + (size_t)pair * 16) * H_DIM;
  const float* wu_base = w13 + ((size_t)e * (2 * I_DIM) + I_DIM + (size_t)pair * 16) * H_DIM;

  const int m  = lane & 15;
  const int hi = lane >> 4;
  const int fro = m * LSTR + 2 * hi;            // fragment row offset (add mt*16*LSTR + kk)

  // precompute staging addresses: X tile = 128x32 floats, 1024 float4 chunks / 256 thr = 4 its
  size_t xgl[4]; int xld[4];
  #pragma unroll
  for (int it = 0; it < 4; ++it) {
    const int idx = tid + 256 * it;
    const int r = idx >> 3, c4 = (idx & 7) * 4;
    xgl[it] = (size_t)stok[r] * H_DIM + c4;
    xld[it] = r * LSTR + c4;
  }
  // weight tiles: 16x32 floats each, 128 chunks / 32 lanes = 4 its (per wave)
  size_t wgl[4]; int wld[4];
  #pragma unroll
  for (int it = 0; it < 4; ++it) {
    const int chunk = lane + 32 * it;
    const int r = chunk >> 3, c4 = (chunk & 7) * 4;
    wgl[it] = (size_t)r * H_DIM + c4;
    wld[it] = r * LSTR + c4;
  }

  v8f accg[MT] = {};
  v8f accu[MT] = {};

  for (int h0 = 0; h0 < H_DIM; h0 += BK) {
    #pragma unroll
    for (int it = 0; it < 4; ++it) {
      cp16_to_lds(&xs[xld[it]], X + xgl[it] + h0);
      cp16_to_lds(&wsg[wave][wld[it]], wg_base + wgl[it] + h0);
      cp16_to_lds(&wsu[wave][wld[it]], wu_base + wgl[it] + h0);
    }
    wait_async_lds();
    __syncthreads();
    #pragma unroll
    for (int kk = 0; kk < BK; kk += 4) {
      const v2f bg = *(const v2f*)&wsg[wave][fro + kk];
      const v2f bu = *(const v2f*)&wsu[wave][fro + kk];
      #pragma unroll
      for (int mt = 0; mt < MT; ++mt) {
        const v2f a = *(const v2f*)&xs[mt * 16 * LSTR + fro + kk];
        accg[mt] = __builtin_amdgcn_wmma_f32_16x16x4_f32(false, a, false, bg, (short)0, accg[mt], false, false);
        accu[mt] = __builtin_amdgcn_wmma_f32_16x16x4_f32(false, a, false, bu, (short)0, accu[mt], false, false);
      }
    }
    __syncthreads();
  }

  #pragma unroll
  for (int mt = 0; mt < MT; ++mt) {
    #pragma unroll
    for (int r = 0; r < 8; ++r) {               // D layout: row = r + 8*hi, col = lane&15
      const float g = accg[mt][r];
      const float u = accu[mt][r];
      const float a = (g / (1.0f + __expf(-g))) * u;   // silu(g) * u
      const int row = mt * 16 + r + 8 * hi;
      act[(size_t)(tb + row) * I_DIM + (size_t)pair * 16 + m] = a;
    }
  }
}

// ---------------- kernel 3: down[entry] = coef * (act_tile @ w2[e]^T) ----------------
// grid.x = entry tiles (128), grid.y = 16 (block covers 8 h-tiles, one per wave -> 128)
__global__ __launch_bounds__(256) void moe_down(const float* __restrict__ act,
                                                const float* __restrict__ w2,
                                                const int* __restrict__ hdr,
                                                const float* __restrict__ ecoef,
                                                float* __restrict__ down) {
  __shared__ float as_[MROWS * LSTR];           // 18432 B
  __shared__ float wsm[8][16 * LSTR];           // 18432 B
  __shared__ float scoef[MROWS];

  const int tile = blockIdx.x;
  const int tb = tile * MROWS;
  if (tb >= hdr[E_EXP]) return;
  int e = 0;
  #pragma unroll
  for (int i = 1; i < E_EXP; ++i) if (tb >= hdr[i]) e = i;

  const int tid = threadIdx.x;
  const int lane = tid & 31;
  const int wave = tid >> 5;
  if (tid < MROWS) scoef[tid] = ecoef[tb + tid];
  __syncthreads();

  const int htile = blockIdx.y * 8 + wave;      // 0..127
  const float* wbase = w2 + ((size_t)e * H_DIM + (size_t)htile * 16) * I_DIM;

  const int m  = lane & 15;
  const int hi = lane >> 4;
  const int fro = m * LSTR + 2 * hi;

  size_t agl[4]; int ald[4];
  #pragma unroll
  for (int it = 0; it < 4; ++it) {
    const int idx = tid + 256 * it;
    const int r = idx >> 3, c4 = (idx & 7) * 4;
    agl[it] = (size_t)(tb + r) * I_DIM + c4;    // act rows are contiguous entries
    ald[it] = r * LSTR + c4;
  }
  size_t wgl[4]; int wld[4];
  #pragma unroll
  for (int it = 0; it < 4; ++it) {
    const int chunk = lane + 32 * it;
    const int r = chunk >> 3, c4 = (chunk & 7) * 4;
    wgl[it] = (size_t)r * I_DIM + c4;
    wld[it] = r * LSTR + c4;
  }

  v8f acc[MT] = {};

  for (int i0 = 0; i0 < I_DIM; i0 += BK) {
    #pragma unroll
    for (int it = 0; it < 4; ++it) {
      cp16_to_lds(&as_[ald[it]], act + agl[it] + i0);
      cp16_to_lds(&wsm[wave][wld[it]], wbase + wgl[it] + i0);
    }
    wait_async_lds();
    __syncthreads();
    #pragma unroll
    for (int kk = 0; kk < BK; kk += 4) {
      const v2f b = *(const v2f*)&wsm[wave][fro + kk];
      #pragma unroll
      for (int mt = 0; mt < MT; ++mt) {
        const v2f a = *(const v2f*)&as_[mt * 16 * LSTR + fro + kk];
        acc[mt] = __builtin_amdgcn_wmma_f32_16x16x4_f32(false, a, false, b, (short)0, acc[mt], false, false);
      }
    }
    __syncthreads();
  }

  #pragma unroll
  for (int mt = 0; mt < MT; ++mt) {
    #pragma unroll
    for (int r = 0; r < 8; ++r) {
      const int row = mt * 16 + r + 8 * hi;
      down[(size_t)(tb + row) * H_DIM + (size_t)htile * 16 + m] = scoef[row] * acc[mt][r];
    }
  }
}

// ---------------- kernel 4: out[t] = down[pos(t,0)] + down[pos(t,1)] ----------------
__global__ __launch_bounds__(256) void moe_reduce(const float* __restrict__ down,
                                                  const int* __restrict__ pos,
                                                  float* __restrict__ out) {
  const size_t idx = (size_t)blockIdx.x * 256 + threadIdx.x;   // over T*H/4
  const int t = (int)(idx >> 9);                               // H/4 = 512 float4 per row
  const int c = (int)(idx & 511);
  const int p0 = pos[2 * t];
  const int p1 = pos[2 * t + 1];
  const v4f* d4 = (const v4f*)down;
  const v4f a = d4[(size_t)p0 * (H_DIM / 4) + c];
  const v4f b = d4[(size_t)p1 * (H_DIM / 4) + c];
  ((v4f*)out)[idx] = a + b;
}

extern "C" void kernel_launch(void* const* d_in, const int* in_sizes, int n_in,
                              void* d_out, int out_size, void* d_ws, size_t ws_size,
                              hipStream_t stream) {
  const float* X   = (const float*)d_in[0];   // [T, H]
  const int*   rt  = (const int*)d_in[1];     // [T, K]
  const float* rw  = (const float*)d_in[2];   // [T, K]
  const float* w13 = (const float*)d_in[3];   // [E, 2I, H]
  const float* w2  = (const float*)d_in[4];   // [E, H, I]
  float* out = (float*)d_out;                 // [T, H]

  char* ws = (char*)d_ws;
  int*   hdr   = (int*)(ws + HDR_OFF);
  int*   pos   = (int*)(ws + POS_OFF);
  int*   etok  = (int*)(ws + ETOK_OFF);
  float* ecoef = (float*)(ws + ECOEF_OFF);
  float* act   = (float*)(ws + ACT_OFF);
  float* down  = (float*)(ws + DOWN_OFF);

  moe_route<<<1, 256, 0, stream>>>(rt, rw, hdr, pos, etok, ecoef);

  const int max_tiles = (T_TOK * K_TOP) / MROWS + E_EXP;       // 264
  dim3 g1(max_tiles, I_DIM / (16 * 8));                        // (264, 8)
  moe_gateup<<<g1, 256, 0, stream>>>(X, w13, hdr, etok, act);

  dim3 g2(max_tiles, H_DIM / (16 * 8));                        // (264, 16)
  moe_down<<<g2, 256, 0, stream>>>(act, w2, hdr, ecoef, down);

  moe_reduce<<<(T_TOK * (H_DIM / 4)) / 256, 256, 0, stream>>>(down, pos, out);
}